// Block_bi_mamba_90211493085241
// MI455X (gfx1250) — compile-verified
//
#include <hip/hip_runtime.h>

#define D_MODEL  256
#define D_INNER  512
#define D_STATE  16
#define D_CONV   4
#define DT_RANK  16
#define XPROJ_OUT 48            // DT_RANK + 2*D_STATE
#define BATCH    2
#define SEQLEN   4096
#define MROWS    (BATCH * SEQLEN)   // 8192
#define CL       256                // scan chunk length
#define NC       (SEQLEN / CL)      // 16 chunks

typedef __attribute__((ext_vector_type(2))) float v2f;
typedef __attribute__((ext_vector_type(8))) float v8f;

__device__ __forceinline__ float silu_f(float x) { return x / (1.f + __expf(-x)); }
__device__ __forceinline__ float softplus_f(float x) {
    return (x > 20.f) ? x : log1pf(__expf(x));
}

// Async global->LDS copy (ASYNCcnt path). LDS offset = low 32 bits of the
// generic pointer (addrspacecast(3->0) = {aperture_hi, lds_offset}).
__device__ __forceinline__ void async_copy_b128(void* lds_dst, const float* src) {
    unsigned loff = (unsigned)(unsigned long long)lds_dst;
    asm volatile("global_load_async_to_lds_b128 %0, %1, off"
                 :: "v"(loff), "v"(src) : "memory");
}
__device__ __forceinline__ void async_wait0() {
    asm volatile("s_wait_asynccnt 0x0" ::: "memory");
}

// ---------------------------------------------------------------------------
// Generic fp32 WMMA GEMM:  OUT[m][n] = epi( sum_k X[m*ldx+k] * W[n*ldw+k] )
// One 16x16 output tile per wave, V_WMMA_F32_16X16X4_F32, K stepped by 4.
// mode 0: plain store | mode 1: softplus(acc + bias[n]) | mode 2: acc + RES[m][n]
// ---------------------------------------------------------------------------
__global__ void k_gemm_wmma(const float* __restrict__ X, int ldx,
                            const float* __restrict__ W, int ldw,
                            float* __restrict__ OUT, int ldo,
                            int tiles_n, int total_tiles, int K, int mode,
                            const float* __restrict__ bias,
                            const float* __restrict__ RES, int ldres)
{
    const int wave = threadIdx.x >> 5;
    const int lane = threadIdx.x & 31;
    const int khi  = lane >> 4;
    const int idx  = lane & 15;

    int tile = blockIdx.x * (blockDim.x >> 5) + wave;
    if (tile >= total_tiles) return;     // wave-uniform
    const int tn = tile % tiles_n;
    const int tm = tile / tiles_n;
    const int m0 = tm << 4, n0 = tn << 4;

    const float* xp = X + (size_t)(m0 + idx) * ldx + 2 * khi;
    const float* wp = W + (size_t)(n0 + idx) * ldw + 2 * khi;

    v8f acc = {0.f, 0.f, 0.f, 0.f, 0.f, 0.f, 0.f, 0.f};
    for (int k = 0; k < K; k += 4) {
        v2f a = *(const v2f*)(xp + k);
        v2f b = *(const v2f*)(wp + k);
        acc = __builtin_amdgcn_wmma_f32_16x16x4_f32(
            false, a, false, b, (short)0, acc, false, false);
    }

    const int n = n0 + idx;
#pragma unroll
    for (int r = 0; r < 8; ++r) {
        const int m = m0 + r + 8 * khi;
        float v = acc[r];
        if (mode == 1)      v = softplus_f(v + bias[n]);
        else if (mode == 2) v = v + RES[(size_t)m * ldres + n];
        OUT[(size_t)m * ldo + n] = v;
    }
}

// ---------------------------------------------------------------------------
// Causal depthwise conv(4) + SiLU. blockIdx.z: 0=fwd, 1=bwd (bwd stored in
// reversed time so all scans run forward in storage order).
// ---------------------------------------------------------------------------
__global__ void k_conv_silu(const float* __restrict__ xz,
                            const float* __restrict__ wf, const float* __restrict__ bf,
                            float* __restrict__ xcf,
                            const float* __restrict__ wb, const float* __restrict__ bb,
                            float* __restrict__ xcb)
{
    const int t = blockIdx.x * blockDim.x + threadIdx.x;
    const int d = t & (D_INNER - 1);
    const int l = (t >> 9) & (SEQLEN - 1);
    const int b = t >> 21;
    const bool back = (blockIdx.z != 0);

    const float* w  = back ? wb : wf;
    const float* bi = back ? bb : bf;
    float*       xc = back ? xcb : xcf;

    float acc = bi[d];
#pragma unroll
    for (int j = 0; j < D_CONV; ++j) {
        const int lj = l - (D_CONV - 1) + j;
        if (lj >= 0) {
            const int lsrc = back ? (SEQLEN - 1 - lj) : lj;
            acc += w[d * D_CONV + j] *
                   xz[((size_t)(b * SEQLEN + lsrc)) * (2 * D_INNER) + d];
        }
    }
    xc[((size_t)(b * SEQLEN + l)) * D_INNER + d] = silu_f(acc);
}

// ---------------------------------------------------------------------------
// Chunked linear-recurrence scan, pass 1: per chunk compute local state
// (h from 0) and decay product P = prod(exp(dt*A)).
// grid = (2 [256-ch blocks], NC, 2*BATCH [b*2+branch]); block = 256.
// ---------------------------------------------------------------------------
__global__ void k_scan_pass1(
    const float* __restrict__ xcf, const float* __restrict__ dtf,
    const float* __restrict__ dblf, const float* __restrict__ Alogf,
    float* __restrict__ Pf, float* __restrict__ Hlocf,
    const float* __restrict__ xcb, const float* __restrict__ dtb,
    const float* __restrict__ dblb, const float* __restrict__ Alogb,
    float* __restrict__ Pb, float* __restrict__ Hlocb)
{
    __shared__ float sB[CL][D_STATE];   // 16KB

    const bool back = (blockIdx.z & 1) != 0;
    const int  b    = blockIdx.z >> 1;
    const int  c    = blockIdx.y;
    const float* xc   = back ? xcb  : xcf;
    const float* dt   = back ? dtb  : dtf;
    const float* dbl  = back ? dblb : dblf;
    const float* Alog = back ? Alogb : Alogf;
    float*       P    = back ? Pb   : Pf;
    float*       Hloc = back ? Hlocb : Hlocf;

    const int d = blockIdx.x * blockDim.x + threadIdx.x;
    const size_t rowbase = (size_t)b * SEQLEN + (size_t)c * CL;

    // async-stage the B columns (CL x 16 floats, 16B segments)
    for (int i = threadIdx.x; i < CL * 4; i += blockDim.x) {
        const int tt = i >> 2, seg = i & 3;
        async_copy_b128(&sB[tt][seg * 4],
                        dbl + (rowbase + tt) * XPROJ_OUT + DT_RANK + seg * 4);
    }

    float A[D_STATE];
#pragma unroll
    for (int n = 0; n < D_STATE; ++n) A[n] = -__expf(Alog[d * D_STATE + n]);

    async_wait0();
    __syncthreads();

    float h[D_STATE], p[D_STATE];
#pragma unroll
    for (int n = 0; n < D_STATE; ++n) { h[n] = 0.f; p[n] = 1.f; }

    for (int t = 0; t < CL; ++t) {
        const size_t row = (rowbase + t) * D_INNER + d;
        const float u   = xc[row];
        const float dtv = dt[row];
#pragma unroll
        for (int n = 0; n < D_STATE; ++n) {
            const float dA = __expf(dtv * A[n]);
            h[n] = dA * h[n] + dtv * sB[t][n] * u;
            p[n] *= dA;
        }
    }

    const size_t o = (((size_t)b * NC + c) * D_INNER + d) * D_STATE;
#pragma unroll
    for (int n = 0; n < D_STATE; ++n) { P[o + n] = p[n]; Hloc[o + n] = h[n]; }
}

// ---------------------------------------------------------------------------
// Pass 2: sequential fix-up across the NC chunks (tiny). One thread per
// (b,d,n); grid = (B*D*N/256, 1, 2 branches).
// ---------------------------------------------------------------------------
__global__ void k_scan_pass2(const float* __restrict__ Pf, const float* __restrict__ Hlf,
                             float* __restrict__ Hsf,
                             const float* __restrict__ Pb, const float* __restrict__ Hlb,
                             float* __restrict__ Hsb)
{
    const bool back = (blockIdx.z != 0);
    const float* P  = back ? Pb  : Pf;
    const float* Hl = back ? Hlb : Hlf;
    float*       Hs = back ? Hsb : Hsf;

    const int i  = blockIdx.x * blockDim.x + threadIdx.x;   // (b*D+d)*N+n
    const int b  = i / (D_INNER * D_STATE);
    const int dn = i % (D_INNER * D_STATE);

    float hs = 0.f;
    for (int c = 0; c < NC; ++c) {
        const size_t o = (size_t)(b * NC + c) * D_INNER * D_STATE + dn;
        Hs[o] = hs;
        hs = Hl[o] + P[o] * hs;
    }
}

// ---------------------------------------------------------------------------
// Pass 3: re-scan each chunk from its corrected start state, emit y.
// ---------------------------------------------------------------------------
__global__ void k_scan_pass3(
    const float* __restrict__ xcf, const float* __restrict__ dtf,
    const float* __restrict__ dblf, const float* __restrict__ Alogf,
    const float* __restrict__ Hsf, const float* __restrict__ Df, float* __restrict__ yf,
    const float* __restrict__ xcb, const float* __restrict__ dtb,
    const float* __restrict__ dblb, const float* __restrict__ Alogb,
    const float* __restrict__ Hsb, const float* __restrict__ Db, float* __restrict__ yb)
{
    __shared__ float sBC[CL][2 * D_STATE];  // 32KB

    const bool back = (blockIdx.z & 1) != 0;
    const int  b    = blockIdx.z >> 1;
    const int  c    = blockIdx.y;
    const float* xc   = back ? xcb  : xcf;
    const float* dt   = back ? dtb  : dtf;
    const float* dbl  = back ? dblb : dblf;
    const float* Alog = back ? Alogb : Alogf;
    const float* Hs   = back ? Hsb  : Hsf;
    const float* Dsk  = back ? Db   : Df;
    float*       y    = back ? yb   : yf;

    const int d = blockIdx.x * blockDim.x + threadIdx.x;
    const size_t rowbase = (size_t)b * SEQLEN + (size_t)c * CL;

    // async-stage B and C (CL x 32 floats, contiguous cols 16..47)
    for (int i = threadIdx.x; i < CL * 8; i += blockDim.x) {
        const int tt = i >> 3, seg = i & 7;
        async_copy_b128(&sBC[tt][seg * 4],
                        dbl + (rowbase + tt) * XPROJ_OUT + DT_RANK + seg * 4);
    }

    float A[D_STATE];
#pragma unroll
    for (int n = 0; n < D_STATE; ++n) A[n] = -__expf(Alog[d * D_STATE + n]);
    const float Dd = Dsk[d];

    float h[D_STATE];
    const size_t o = (((size_t)b * NC + c) * D_INNER + d) * D_STATE;
#pragma unroll
    for (int n = 0; n < D_STATE; ++n) h[n] = Hs[o + n];

    async_wait0();
    __syncthreads();

    for (int t = 0; t < CL; ++t) {
        const size_t row = (rowbase + t) * D_INNER + d;
        const float u   = xc[row];
        const float dtv = dt[row];
        float acc = 0.f;
#pragma unroll
        for (int n = 0; n < D_STATE; ++n) {
            const float dA = __expf(dtv * A[n]);
            h[n] = dA * h[n] + dtv * sBC[t][n] * u;
            acc += h[n] * sBC[t][D_STATE + n];
        }
        y[row] = acc + u * Dd;
    }
}

// ---------------------------------------------------------------------------
// y = (y_f + reverse(y_b)) * silu(z); in-place into y_f.
// ---------------------------------------------------------------------------
__global__ void k_combine(const float* __restrict__ yfin, const float* __restrict__ ybin,
                          const float* __restrict__ xz, float* __restrict__ yc)
{
    const int t = blockIdx.x * blockDim.x + threadIdx.x;
    const int d = t & (D_INNER - 1);
    const int l = (t >> 9) & (SEQLEN - 1);
    const int b = t >> 21;
    const size_t row  = (size_t)(b * SEQLEN + l);
    const size_t rrow = (size_t)(b * SEQLEN + (SEQLEN - 1 - l));
    const float z = xz[row * (2 * D_INNER) + D_INNER + d];
    const float v = yfin[row * D_INNER + d] + ybin[rrow * D_INNER + d];
    yc[row * D_INNER + d] = v * silu_f(z);
}

// ---------------------------------------------------------------------------
extern "C" void kernel_launch(void* const* d_in, const int* in_sizes, int n_in,
                              void* d_out, int out_size, void* d_ws, size_t ws_size,
                              hipStream_t stream)
{
    const float* x          = (const float*)d_in[0];
    const float* in_proj_w  = (const float*)d_in[1];
    const float* conv_w_f   = (const float*)d_in[2];
    const float* conv_b_f   = (const float*)d_in[3];
    const float* xproj_w_f  = (const float*)d_in[4];
    const float* dtproj_w_f = (const float*)d_in[5];
    const float* dtproj_b_f = (const float*)d_in[6];
    const float* A_log_f    = (const float*)d_in[7];
    const float* D_f        = (const float*)d_in[8];
    const float* conv_w_b   = (const float*)d_in[9];
    const float* conv_b_b   = (const float*)d_in[10];
    const float* xproj_w_b  = (const float*)d_in[11];
    const float* dtproj_w_b = (const float*)d_in[12];
    const float* dtproj_b_b = (const float*)d_in[13];
    const float* A_log_b    = (const float*)d_in[14];
    const float* D_b        = (const float*)d_in[15];
    const float* out_proj_w = (const float*)d_in[16];
    float* out = (float*)d_out;

    const size_t STATE_SZ = (size_t)BATCH * NC * D_INNER * D_STATE; // 262144

    float* ws   = (float*)d_ws;
    float* xz   = ws;                                       // 8192*1024
    float* xcf  = xz  + (size_t)MROWS * 2 * D_INNER;
    float* xcb  = xcf + (size_t)MROWS * D_INNER;
    float* dblf = xcb + (size_t)MROWS * D_INNER;
    float* dblb = dblf + (size_t)MROWS * XPROJ_OUT;
    float* dtfb = dblb + (size_t)MROWS * XPROJ_OUT;
    float* dtbb = dtfb + (size_t)MROWS * D_INNER;
    float* yfb  = dtbb + (size_t)MROWS * D_INNER;
    float* ybb  = yfb + (size_t)MROWS * D_INNER;
    float* Pf   = ybb + (size_t)MROWS * D_INNER;
    float* Hlf  = Pf  + STATE_SZ;
    float* Hsf  = Hlf + STATE_SZ;
    float* Pb   = Hsf + STATE_SZ;
    float* Hlb  = Pb  + STATE_SZ;
    float* Hsb  = Hlb + STATE_SZ;

    const dim3 blk(256);

    // 1) in_proj: xz = x @ in_proj_w^T   (M=8192, N=1024, K=256)
    k_gemm_wmma<<<(512 * 64) / 8, blk, 0, stream>>>(
        x, D_MODEL, in_proj_w, D_MODEL, xz, 2 * D_INNER,
        64, 512 * 64, D_MODEL, 0, nullptr, nullptr, 0);

    // 2) conv + silu, both directions
    k_conv_silu<<<dim3(16384, 1, 2), blk, 0, stream>>>(
        xz, conv_w_f, conv_b_f, xcf, conv_w_b, conv_b_b, xcb);

    // 3) xproj (M=8192, N=48, K=512)
    k_gemm_wmma<<<(512 * 3) / 8, blk, 0, stream>>>(
        xcf, D_INNER, xproj_w_f, D_INNER, dblf, XPROJ_OUT,
        3, 512 * 3, D_INNER, 0, nullptr, nullptr, 0);
    k_gemm_wmma<<<(512 * 3) / 8, blk, 0, stream>>>(
        xcb, D_INNER, xproj_w_b, D_INNER, dblb, XPROJ_OUT,
        3, 512 * 3, D_INNER, 0, nullptr, nullptr, 0);

    // 4) dt = softplus(dtr @ dtproj_w^T + bias)   (M=8192, N=512, K=16)
    k_gemm_wmma<<<(512 * 32) / 8, blk, 0, stream>>>(
        dblf, XPROJ_OUT, dtproj_w_f, DT_RANK, dtfb, D_INNER,
        32, 512 * 32, DT_RANK, 1, dtproj_b_f, nullptr, 0);
    k_gemm_wmma<<<(512 * 32) / 8, blk, 0, stream>>>(
        dblb, XPROJ_OUT, dtproj_w_b, DT_RANK, dtbb, D_INNER,
        32, 512 * 32, DT_RANK, 1, dtproj_b_b, nullptr, 0);

    // 5) chunked parallel scan (both branches via grid.z)
    k_scan_pass1<<<dim3(2, NC, 2 * BATCH), blk, 0, stream>>>(
        xcf, dtfb, dblf, A_log_f, Pf, Hlf,
        xcb, dtbb, dblb, A_log_b, Pb, Hlb);
    k_scan_pass2<<<dim3((BATCH * D_INNER * D_STATE) / 256, 1, 2), blk, 0, stream>>>(
        Pf, Hlf, Hsf, Pb, Hlb, Hsb);
    k_scan_pass3<<<dim3(2, NC, 2 * BATCH), blk, 0, stream>>>(
        xcf, dtfb, dblf, A_log_f, Hsf, D_f, yfb,
        xcb, dtbb, dblb, A_log_b, Hsb, D_b, ybb);

    // 6) combine + gate (in-place into yfb)
    k_combine<<<16384, blk, 0, stream>>>(yfb, ybb, xz, yfb);

    // 7) out_proj + residual (M=8192, N=256, K=512)
    k_gemm_wmma<<<(512 * 16) / 8, blk, 0, stream>>>(
        yfb, D_INNER, out_proj_w, D_INNER, out, D_MODEL,
        16, 512 * 16, D_INNER, 2, nullptr, x, D_MODEL);
}